// LightGCNLayer_55929064129243
// MI455X (gfx1250) — compile-verified
//
#include <hip/hip_runtime.h>
#include <stdint.h>

#define DFEAT 64   // feature dim
#define TPE   16   // threads per edge (16 x float4 = 64 floats)
#define EPB   16   // edges per 256-thread block iteration

#ifndef __has_builtin
#define __has_builtin(x) 0
#endif

#if __has_builtin(__builtin_amdgcn_global_load_async_to_lds_b128) && \
    __has_builtin(__builtin_amdgcn_s_wait_asynccnt)
#define USE_ASYNC 1
#else
#define USE_ASYNC 0
#endif

__device__ __forceinline__ void atomic_add_f32(float* p, float v) {
  // native L2 atomic, no return value -> STOREcnt-tracked global_atomic_add_f32
  __hip_atomic_fetch_add(p, v, __ATOMIC_RELAXED, __HIP_MEMORY_SCOPE_AGENT);
}

__global__ void __launch_bounds__(256)
zero_kernel(float* out, long n) {
  long i = (long)blockIdx.x * blockDim.x + threadIdx.x;
  long n4 = n >> 2;
  if (i < n4) reinterpret_cast<float4*>(out)[i] = make_float4(0.f, 0.f, 0.f, 0.f);
  if (i == 0) {
    for (long k = n4 << 2; k < n; ++k) out[k] = 0.f;
  }
}

// ---------------- fallback: plain vectorized gather + atomic scatter ----------------
__global__ void __launch_bounds__(256)
scatter_plain(const float* __restrict__ feat, const float* __restrict__ norm,
              const int* __restrict__ src, const int* __restrict__ dst,
              float* __restrict__ out, int n_edges) {
  const int sub = threadIdx.x & (TPE - 1);            // which float4 of the row
  long tid = (long)blockIdx.x * blockDim.x + threadIdx.x;
  long e = tid >> 4;
  if (e >= n_edges) return;
  const int   s = src[e];
  const int   d = dst[e];
  const float w = norm[e];
  const float4 v = *reinterpret_cast<const float4*>(feat + (size_t)s * DFEAT + sub * 4);
  float* o = out + (size_t)d * DFEAT + sub * 4;
  atomic_add_f32(o + 0, w * v.x);
  atomic_add_f32(o + 1, w * v.y);
  atomic_add_f32(o + 2, w * v.z);
  atomic_add_f32(o + 3, w * v.w);
}

// ---------------- CDNA5 path: async DMA gather to LDS, double buffered ----------------
#if USE_ASYNC
typedef int v4i __attribute__((ext_vector_type(4)));
typedef __attribute__((address_space(1))) v4i* gptr_t;  // global int4*
typedef __attribute__((address_space(3))) v4i* lptr_t;  // LDS int4*

__device__ __forceinline__ void async_copy_b128(const float* gp, const float* lp) {
  // generic == global address for AS1; low 32 bits of a generic shared
  // address are the LDS byte offset (LDS aperture: addr[31:0] == offset)
  __builtin_amdgcn_global_load_async_to_lds_b128(
      (gptr_t)(uintptr_t)gp,
      (lptr_t)(uint32_t)(uintptr_t)lp,
      0, 0);
}

__global__ void __launch_bounds__(256)
scatter_async(const float* __restrict__ feat, const float* __restrict__ norm,
              const int* __restrict__ src, const int* __restrict__ dst,
              float* __restrict__ out, int n_edges) {
  // 2 buffers x 256 lanes x 16B = 8 KB; each wave touches only its own 32x16B slice,
  // so no barriers are needed -- only s_wait_asynccnt.
  __shared__ float lds[2][256 * 4];
  const int t    = threadIdx.x;
  const int sub  = t & (TPE - 1);
  const int eloc = t >> 4;
  const int nchunks = (n_edges + EPB - 1) / EPB;
  const int cstride = gridDim.x;

  int cur = blockIdx.x;
  if (cur >= nchunks) return;

  // prologue: DMA chunk `cur` into buffer 0 (clamped index keeps EXEC full)
  {
    int e = cur * EPB + eloc;
    int s = (e < n_edges) ? src[e] : 0;
    async_copy_b128(feat + (size_t)s * DFEAT + sub * 4, &lds[0][t * 4]);
  }

  int buf = 0;
  for (;;) {
    const int nxt = cur + cstride;
    if (nxt < nchunks) {
      // issue next chunk into the other buffer, then wait for current one
      int e = nxt * EPB + eloc;
      int s = (e < n_edges) ? src[e] : 0;
      async_copy_b128(feat + (size_t)s * DFEAT + sub * 4, &lds[buf ^ 1][t * 4]);
      __builtin_amdgcn_s_wait_asynccnt(1);
    } else {
      __builtin_amdgcn_s_wait_asynccnt(0);
    }
    __asm__ volatile("" ::: "memory");  // keep LDS reads below the async wait

    // process chunk `cur` out of lds[buf]
    int e = cur * EPB + eloc;
    if (e < n_edges) {
      const float w = norm[e];
      const int   d = dst[e];
      const float4 v = *reinterpret_cast<const float4*>(&lds[buf][t * 4]);
      float* o = out + (size_t)d * DFEAT + sub * 4;
      atomic_add_f32(o + 0, w * v.x);
      atomic_add_f32(o + 1, w * v.y);
      atomic_add_f32(o + 2, w * v.z);
      atomic_add_f32(o + 3, w * v.w);
    }

    if (nxt >= nchunks) break;
    cur = nxt;
    buf ^= 1;
  }
}
#endif  // USE_ASYNC

extern "C" void kernel_launch(void* const* d_in, const int* in_sizes, int n_in,
                              void* d_out, int out_size, void* d_ws, size_t ws_size,
                              hipStream_t stream) {
  const float* user_feat = (const float*)d_in[0];
  const float* item_feat = (const float*)d_in[1];
  const float* norm_ui   = (const float*)d_in[2];
  const float* norm_iu   = (const float*)d_in[3];
  const int*   ui_src    = (const int*)d_in[4];
  const int*   ui_dst    = (const int*)d_in[5];
  const int*   iu_src    = (const int*)d_in[6];
  const int*   iu_dst    = (const int*)d_in[7];

  const int n_users = in_sizes[0] / DFEAT;       // 100000
  const int n_edges = in_sizes[4];               // 2000000

  float* h_user = (float*)d_out;                                    // [n_users, 64]
  float* h_item = (float*)d_out + (size_t)n_users * DFEAT;          // [n_items, 64]

  // zero the accumulators (harness poisons d_out)
  {
    long n = (long)out_size;
    long n4 = n >> 2;
    int blocks = (int)((n4 + 255) / 256);
    if (blocks < 1) blocks = 1;
    zero_kernel<<<blocks, 256, 0, stream>>>((float*)d_out, n);
  }

#if USE_ASYNC
  {
    int nchunks = (n_edges + EPB - 1) / EPB;
    int blocks = nchunks < 8192 ? nchunks : 8192;
    // relation (user -ui-> item): gather user rows, scatter-sum into h_item
    scatter_async<<<blocks, 256, 0, stream>>>(user_feat, norm_ui, ui_src, ui_dst, h_item, n_edges);
    // relation (item -iu-> user): gather item rows, scatter-sum into h_user
    scatter_async<<<blocks, 256, 0, stream>>>(item_feat, norm_iu, iu_src, iu_dst, h_user, n_edges);
  }
#else
  {
    long tthreads = (long)n_edges * TPE;
    int blocks = (int)((tthreads + 255) / 256);
    scatter_plain<<<blocks, 256, 0, stream>>>(user_feat, norm_ui, ui_src, ui_dst, h_item, n_edges);
    scatter_plain<<<blocks, 256, 0, stream>>>(item_feat, norm_iu, iu_src, iu_dst, h_user, n_edges);
  }
#endif
}